// QuantumClassifier_22565758173818
// MI455X (gfx1250) — compile-verified
//
#include <hip/hip_runtime.h>

// ---------------------------------------------------------------------------
// 14-qubit state-vector simulator, one sample per workgroup, state in LDS.
// Four single-qubit gates are fused into a 16x16 complex unitary and applied
// to the (16 x 1024)-reshaped state with V_WMMA_F32_16X16X4_F32.
// CNOT entangler layers are GF(2)-linear index permutations; they are folded
// into a lazily-maintained logical->physical bit-linear map (14 column masks)
// instead of moving any data: one CNOT == one XOR on the mask table.
// ---------------------------------------------------------------------------

typedef __attribute__((ext_vector_type(2))) float v2f;
typedef __attribute__((ext_vector_type(8))) float v8f;

#define NQ      14
#define NSTATE  16384          // 2^14
#define NLAYERS 6
#define BLOCK   256            // 8 wave32s

// Qubit q lives at LOGICAL bit position (13 - q) of the amplitude index
// (qubit0 = MSB, matching the reference's row-major (2,)*14 layout).
// Groups of 4 qubits fused into one 16x16 unitary. Group 3 reuses qubits
// 10,11 as identity fillers (only 12,13 are active there).
__device__ const int GQ_[4][4]   = {{0,1,2,3},{4,5,6,7},{8,9,10,11},{10,11,12,13}};
__device__ const int GPOS[4][4]  = {{13,12,11,10},{9,8,7,6},{5,4,3,2},{3,2,1,0}};
__device__ const int GLOW[4]     = {10, 6, 2, 0};     // lowest logical position of group
__device__ const int GACT[4]     = {0xF,0xF,0xF,0xC}; // active-slot mask (group3: 12,13)

// Build the 16x16 complex unitary for group g.  layer < 0 => per-sample RY
// encoding gates; layer >= 0 => shared Rot(phi,theta,omega) gates.
static __device__ __forceinline__ void build_group(float* gre, float* gim, float* us,
                                                   int g, int layer,
                                                   const float* __restrict__ x,
                                                   const float* __restrict__ w,
                                                   int b, int tid) {
    if (tid < 4) {
        const int t = tid;
        const int q = GQ_[g][t];
        float u00r = 1.f, u00i = 0.f, u01r = 0.f, u01i = 0.f;
        float u10r = 0.f, u10i = 0.f, u11r = 1.f, u11i = 0.f;
        if ((GACT[g] >> t) & 1) {
            if (layer < 0) {
                // RY(x*pi): theta/2 = x*pi/2
                float h  = x[b * NQ + q] * 1.57079632679489662f;
                float ch = cosf(h), sh = sinf(h);
                u00r =  ch; u01r = -sh; u10r = sh; u11r = ch;
            } else {
                const float* wp = w + ((layer * NQ) + q) * 3;
                float phi = wp[0], th = wp[1], om = wp[2];
                float ct = cosf(0.5f * th), st = sinf(0.5f * th);
                float a  = 0.5f * (phi + om);
                float d  = 0.5f * (phi - om);
                float ca = cosf(a), sa = sinf(a);
                float cd = cosf(d), sd = sinf(d);
                // Rot = RZ(om) RY(th) RZ(phi)
                u00r =  ca * ct;  u00i = -sa * ct;   //  e^{-i a} ct
                u01r = -cd * st;  u01i = -sd * st;   // -e^{+i d} st
                u10r =  cd * st;  u10i = -sd * st;   //  e^{-i d} st
                u11r =  ca * ct;  u11i =  sa * ct;   //  e^{+i a} ct
            }
        }
        float* up = us + t * 8;
        up[0] = u00r; up[1] = u00i; up[2] = u01r; up[3] = u01i;
        up[4] = u10r; up[5] = u10i; up[6] = u11r; up[7] = u11i;
    }
    __syncthreads();
    // 256 threads -> one G entry each: G[i][j] = prod_t U_t[bit_t(i)][bit_t(j)]
    const int i = tid >> 4, j = tid & 15;
    float gr = 1.f, gi = 0.f;
#pragma unroll
    for (int t = 0; t < 4; ++t) {
        const int bi = (i >> t) & 1, bj = (j >> t) & 1;
        const float* up = us + t * 8 + (bi * 2 + bj) * 2;
        const float ur = up[0], ui = up[1];
        const float nr = gr * ur - gi * ui;
        const float ni = gr * ui + gi * ur;
        gr = nr; gi = ni;
    }
    gre[tid] = gr;
    gim[tid] = gi;
}

// Apply the 16x16 complex unitary G to the state viewed as (16 group rows) x
// (1024 rest columns) using f32 WMMA.  Addresses go through the cumulative
// GF(2)-linear logical->physical map (cmask).  Each wave owns 8 of the 64
// 16-column tiles; logical columns are disjoint across lanes and the map is
// invertible, so physical addresses are disjoint too -> in-place is safe.
static __device__ __forceinline__ void apply_group(float* sre, float* sim,
                                                   const float* gre, const float* gim,
                                                   const int* cmask,
                                                   int g, int tid) {
    const int lane = tid & 31;
    const int wv   = tid >> 5;     // wave 0..7
    const int hi   = lane >> 4;    // lane half
    const int lo   = lane & 15;
    const int p0 = GPOS[g][0], p1 = GPOS[g][1], p2 = GPOS[g][2], p3 = GPOS[g][3];
    const int P  = GLOW[g];

    // physical masks of the 4 group bits and of all 14 logical bits
    const int pm0 = cmask[p0], pm1 = cmask[p1], pm2 = cmask[p2], pm3 = cmask[p3];
    int cm[NQ];
#pragma unroll
    for (int p = 0; p < NQ; ++p) cm[p] = cmask[p];

    // A fragments (G matrix): A[m][k], lane<16 holds K0/K1, lane>=16 K2/K3.
    v2f Ar[4], Aim[4], Ain[4];
#pragma unroll
    for (int c = 0; c < 4; ++c) {
#pragma unroll
        for (int vv = 0; vv < 2; ++vv) {
            const int jj = 4 * c + 2 * hi + vv;
            const float r  = gre[lo * 16 + jj];
            const float im = gim[lo * 16 + jj];
            Ar[c][vv]  = r;
            Aim[c][vv] = im;
            Ain[c][vv] = -im;   // f32 WMMA has no A-neg modifier -> pre-negate
        }
    }

#pragma unroll 1
    for (int tile = wv; tile < 64; tile += 8) {
        const int kcol  = tile * 16 + lo;   // logical rest-column index
        // deposit the 10 rest bits around the contiguous group range [P, P+4)
        const int lbase = ((kcol >> P) << (P + 4)) | (kcol & ((1 << P) - 1));
        // physical base address through the linear map
        int pbase = 0;
#pragma unroll
        for (int p = 0; p < NQ; ++p)
            pbase ^= ((lbase >> p) & 1) ? cm[p] : 0;

        // B fragments (state): B[k][n], row k = 4c + 2v + hi, col n = lo.
        v2f Br[4], Bi[4];
#pragma unroll
        for (int c = 0; c < 4; ++c) {
#pragma unroll
            for (int vv = 0; vv < 2; ++vv) {
                const int jj  = 4 * c + 2 * vv + hi;
                const int off = ((jj & 1)        ? pm0 : 0) ^
                                (((jj >> 1) & 1) ? pm1 : 0) ^
                                (((jj >> 2) & 1) ? pm2 : 0) ^
                                (((jj >> 3) & 1) ? pm3 : 0);
                Br[c][vv] = sre[pbase ^ off];
                Bi[c][vv] = sim[pbase ^ off];
            }
        }

        v8f Cr = {};
        v8f Ci = {};
#pragma unroll
        for (int c = 0; c < 4; ++c) {
            // Cr += Gr*Sr - Gi*Si ;  Ci += Gr*Si + Gi*Sr
            Cr = __builtin_amdgcn_wmma_f32_16x16x4_f32(false, Ar[c],  false, Br[c], (short)0, Cr, false, false);
            Cr = __builtin_amdgcn_wmma_f32_16x16x4_f32(false, Ain[c], false, Bi[c], (short)0, Cr, false, false);
            Ci = __builtin_amdgcn_wmma_f32_16x16x4_f32(false, Ar[c],  false, Bi[c], (short)0, Ci, false, false);
            Ci = __builtin_amdgcn_wmma_f32_16x16x4_f32(false, Aim[c], false, Br[c], (short)0, Ci, false, false);
        }

        // C layout: VGPR r -> M = r + 8*hi, N = lo (same column -> same pbase).
#pragma unroll
        for (int r = 0; r < 8; ++r) {
            const int m   = r + 8 * hi;
            const int off = ((m & 1)        ? pm0 : 0) ^
                            (((m >> 1) & 1) ? pm1 : 0) ^
                            (((m >> 2) & 1) ? pm2 : 0) ^
                            (((m >> 3) & 1) ? pm3 : 0);
            sre[pbase ^ off] = Cr[r];
            sim[pbase ^ off] = Ci[r];
        }
    }
}

__global__ void __launch_bounds__(BLOCK, 2)
qc14_wmma_kernel(const float* __restrict__ x, const float* __restrict__ w,
                 float* __restrict__ out) {
    extern __shared__ float smem[];
    float* sre = smem;                 // 16384
    float* sim = smem + NSTATE;        // 16384
    float* gre = sim + NSTATE;         // 256
    float* gim = gre + 256;            // 256
    float* us  = gim + 256;            // 32
    float* red = us + 32;              // 512
    int*   cmask = (int*)(red + 2 * BLOCK); // 16 ints (14 used)

    const int tid = threadIdx.x;
    const int b   = blockIdx.x;

    // |0...0>  and identity logical->physical map
    for (int a = tid; a < NSTATE; a += BLOCK) { sre[a] = 0.f; sim[a] = 0.f; }
    if (tid == 0) sre[0] = 1.f;
    if (tid < NQ) cmask[tid] = 1 << tid;
    __syncthreads();

    // ---- angle encoding (per-sample RY gates, fused 4 at a time) ----
    for (int g = 0; g < 4; ++g) {
        build_group(gre, gim, us, g, -1, x, w, b, tid);
        __syncthreads();
        apply_group(sre, sim, gre, gim, cmask, g, tid);
        __syncthreads();
    }

    // ---- StronglyEntanglingLayers ----
    for (int l = 0; l < NLAYERS; ++l) {
        for (int g = 0; g < 4; ++g) {
            build_group(gre, gim, us, g, l, x, w, b, tid);
            __syncthreads();
            apply_group(sre, sim, gre, gim, cmask, g, tid);
            __syncthreads();
        }
        // CNOT ring with range r: pure GF(2)-linear index permutation.
        // Composing CNOT(c,t):  cmask[pos(c)] ^= cmask[pos(t)]   (in order).
        if (tid == 0) {
            const int r = (l % (NQ - 1)) + 1;
            for (int q = 0; q < NQ; ++q) {
                const int pc = 13 - q;
                const int pt = 13 - ((q + r) % NQ);
                cmask[pc] ^= cmask[pt];
            }
        }
        __syncthreads();
    }

    // ---- <Z> on qubits 0 (logical bit13) and 1 (logical bit12) ----
    int cm[NQ];
#pragma unroll
    for (int p = 0; p < NQ; ++p) cm[p] = cmask[p];
    float z0 = 0.f, z1 = 0.f;
    for (int a = tid; a < NSTATE; a += BLOCK) {
        int phys = 0;
#pragma unroll
        for (int p = 0; p < NQ; ++p)
            phys ^= ((a >> p) & 1) ? cm[p] : 0;
        const float pr = sre[phys] * sre[phys] + sim[phys] * sim[phys];
        z0 += (a & 0x2000) ? -pr : pr;
        z1 += (a & 0x1000) ? -pr : pr;
    }
    red[tid]         = z0;
    red[BLOCK + tid] = z1;
    __syncthreads();
    for (int s = BLOCK / 2; s > 0; s >>= 1) {
        if (tid < s) {
            red[tid]         += red[tid + s];
            red[BLOCK + tid] += red[BLOCK + tid + s];
        }
        __syncthreads();
    }
    if (tid == 0) {
        out[b * 2 + 0] = red[0];
        out[b * 2 + 1] = red[BLOCK];
    }
}

extern "C" void kernel_launch(void* const* d_in, const int* in_sizes, int n_in,
                              void* d_out, int out_size, void* d_ws, size_t ws_size,
                              hipStream_t stream) {
    (void)n_in; (void)out_size; (void)d_ws; (void)ws_size;
    const float* x = (const float*)d_in[0];   // (B, 14) f32
    const float* w = (const float*)d_in[1];   // (6, 14, 3) f32
    float* out     = (float*)d_out;           // (B, 2) f32

    const int B = in_sizes[0] / NQ;           // 4096
    const size_t smem_bytes =
        (size_t)(2 * NSTATE + 256 + 256 + 32 + 2 * BLOCK + 16) * sizeof(float);

    hipFuncSetAttribute((const void*)qc14_wmma_kernel,
                        hipFuncAttributeMaxDynamicSharedMemorySize,
                        (int)smem_bytes);

    qc14_wmma_kernel<<<B, BLOCK, smem_bytes, stream>>>(x, w, out);
}